// GinDecoderLayer_23450521436278
// MI455X (gfx1250) — compile-verified
//
#include <hip/hip_runtime.h>
#include <hip/hip_bf16.h>

typedef __attribute__((ext_vector_type(2))) float v2f;
typedef __attribute__((ext_vector_type(8))) float v8f;
typedef __attribute__((__vector_size__(4 * sizeof(int)))) int v4i;

#define BINS 1024
#define DIMS 256
#define TSTRIDE 68                        // 64 cols + 4 pad: 16B-aligned rows, conflict-free A-frags
#define CHUNK_FLOATS (16 * TSTRIDE)       // one 16x64 staged chunk  = 1088 floats
#define TILE_FLOATS  (4 * CHUNK_FLOATS)   // one full 16x256 tile    = 4352 floats
#define WPB 8                             // waves per block
#define TPB 256
#define GRID_MAIN 1024

// Unified dynamic-LDS layout (floats): tiles | sum | cnt | w | seg
#define SMEM_FLOATS (WPB * 2 * TILE_FLOATS + 2 * BINS + DIMS + 16 * WPB)

#if defined(__has_builtin)
#if __has_builtin(__builtin_amdgcn_global_load_async_to_lds_b128) && \
    __has_builtin(__builtin_amdgcn_s_wait_asynccnt)
#define USE_ASYNC 1
#endif
#endif

__global__ void gin_zero_ws(float* ws, int n) {
    int i = blockIdx.x * blockDim.x + threadIdx.x;
    if (i < n) ws[i] = 0.0f;
}

#if defined(USE_ASYNC) && defined(__HIP_DEVICE_COMPILE__)
typedef __attribute__((address_space(1))) v4i* async_gptr;
typedef __attribute__((address_space(3))) v4i* async_lptr;

// Issue a full 16x256 f32 tile as 32 async b128 copies per lane: HBM -> LDS,
// landing in padded-chunk layout (4 chunks of 16 x TSTRIDE). No VGPR staging.
__device__ __forceinline__ void issue_tile_async(const float* __restrict__ g,
                                                 float* lbuf, int lane) {
#pragma unroll
    for (int j = 0; j < 32; ++j) {
        int e4 = j * 32 + lane;           // float4 index in 16x256 tile (row-major)
        int r  = e4 >> 6;                 // 64 float4 per row
        int c4 = e4 & 63;
        const float* gp = g + r * DIMS + c4 * 4;
        float* lp = lbuf + (c4 >> 4) * CHUNK_FLOATS + r * TSTRIDE + (c4 & 15) * 4;
        __builtin_amdgcn_global_load_async_to_lds_b128(
            (async_gptr)(void*)gp, (async_lptr)(void*)lp, 0, 0);
    }
}
#endif

__global__ __launch_bounds__(TPB)
void gin_seg_dot(const float* __restrict__ nodes,
                 const int*   __restrict__ gidx,
                 const float* __restrict__ W,
                 float* __restrict__ gsum,
                 float* __restrict__ gcnt,
                 int N) {
    extern __shared__ float smem[];
    float* s_tile = smem;                            // [WPB][2][TILE_FLOATS]
    float* s_sum  = smem + WPB * 2 * TILE_FLOATS;    // [BINS]
    float* s_cnt  = s_sum + BINS;                    // [BINS]
    float* s_w    = s_cnt + BINS;                    // [DIMS]
    int*   s_seg  = (int*)(s_w + DIMS);              // [WPB*16]

    const int tid = threadIdx.x;
    for (int i = tid; i < BINS; i += TPB) { s_sum[i] = 0.0f; s_cnt[i] = 0.0f; }
    for (int i = tid; i < DIMS; i += TPB) s_w[i] = W[i];
    __syncthreads();

    const int lane = tid & 31;
    const int wv   = tid >> 5;
    int*   segbuf = s_seg + wv * 16;
    const int row  = lane & 15;          // A-frag: row this lane holds
    const int cofs = (lane >> 4) << 1;   // lanes 16-31 hold K+2,K+3

    const int ntiles     = N >> 4;
    const int waveGlobal = blockIdx.x * WPB + wv;
    const int waveStride = gridDim.x * WPB;

#if defined(USE_ASYNC) && defined(__HIP_DEVICE_COMPILE__)
    // ---- Async path: double-buffered full tiles, ASYNCcnt waits ----
    float* b0 = s_tile + wv * 2 * TILE_FLOATS;
    float* b1 = b0 + TILE_FLOATS;

    int tile = waveGlobal;
    if (tile < ntiles)
        issue_tile_async(nodes + (long long)tile * (16 * DIMS), b0, lane);
    int cur = 0;
    for (; tile < ntiles; tile += waveStride) {
        const int nxt = tile + waveStride;
        float* cbuf = cur ? b1 : b0;
        float* nbuf = cur ? b0 : b1;
        if (nxt < ntiles) {
            issue_tile_async(nodes + (long long)nxt * (16 * DIMS), nbuf, lane);
            // in-order completion: <=32 outstanding => current tile fully in LDS
            __builtin_amdgcn_s_wait_asynccnt(32);
        } else {
            __builtin_amdgcn_s_wait_asynccnt(0);
        }

        if (lane < 16) {
            int s = gidx[tile * 16 + lane];
            segbuf[lane] = s;
            atomicAdd(&s_cnt[s], 1.0f);              // ds_add_f32
        }

        v8f acc = {0.f, 0.f, 0.f, 0.f, 0.f, 0.f, 0.f, 0.f};
#pragma unroll
        for (int kc = 0; kc < 4; ++kc) {
            const float* cb = cbuf + kc * CHUNK_FLOATS;
#pragma unroll
            for (int k = 0; k < 16; ++k) {
                v2f a = *(const v2f*)(cb + row * TSTRIDE + k * 4 + cofs);
                v2f b = *(const v2f*)(s_w + kc * 64 + k * 4 + cofs);
                acc = __builtin_amdgcn_wmma_f32_16x16x4_f32(
                        false, a, false, b, (short)0, acc, false, false);
            }
        }
        // D layout: VGPR j -> (lane 0: M=j, N=0), (lane 16: M=j+8, N=0)
        if (lane == 0 || lane == 16) {
            const int mb = (lane == 0) ? 0 : 8;
#pragma unroll
            for (int j = 0; j < 8; ++j)
                atomicAdd(&s_sum[segbuf[mb + j]], acc[j]);
        }
        cur ^= 1;
    }
#else
    // ---- Fallback: synchronous b128 -> VGPR -> LDS staging ----
    float* tbuf = s_tile + wv * 2 * TILE_FLOATS;     // use first chunk of our region
    for (int tile = waveGlobal; tile < ntiles; tile += waveStride) {
        if (tile + waveStride < ntiles)
            __builtin_prefetch(nodes + (size_t)(tile + waveStride) * (16 * DIMS), 0, 1);
        const long long base = (long long)tile * (16 * DIMS);
        if (lane < 16) {
            int s = gidx[tile * 16 + lane];
            segbuf[lane] = s;
            atomicAdd(&s_cnt[s], 1.0f);
        }
        v8f acc = {0.f, 0.f, 0.f, 0.f, 0.f, 0.f, 0.f, 0.f};
#pragma unroll
        for (int kc = 0; kc < 4; ++kc) {
#pragma unroll
            for (int j = 0; j < 8; ++j) {
                int e4 = j * 32 + lane;
                int r  = e4 >> 4;
                int c4 = e4 & 15;
                const float4 v = *(const float4*)(nodes + base + (long long)r * DIMS + kc * 64 + c4 * 4);
                *(float4*)(tbuf + r * TSTRIDE + c4 * 4) = v;
            }
#pragma unroll
            for (int k = 0; k < 16; ++k) {
                v2f a = *(const v2f*)(tbuf + row * TSTRIDE + k * 4 + cofs);
                v2f b = *(const v2f*)(s_w + kc * 64 + k * 4 + cofs);
                acc = __builtin_amdgcn_wmma_f32_16x16x4_f32(
                        false, a, false, b, (short)0, acc, false, false);
            }
        }
        if (lane == 0 || lane == 16) {
            const int mb = (lane == 0) ? 0 : 8;
#pragma unroll
            for (int j = 0; j < 8; ++j)
                atomicAdd(&s_sum[segbuf[mb + j]], acc[j]);
        }
    }
#endif

    // Scalar tail for N % 16 rows (none for N = 1M, kept for generality)
    if (blockIdx.x == 0 && wv == 0) {
        for (int r = ntiles * 16; r < N; ++r) {
            float p = 0.0f;
            for (int c = lane; c < DIMS; c += 32)
                p += nodes[(long long)r * DIMS + c] * s_w[c];
#pragma unroll
            for (int off = 16; off > 0; off >>= 1)
                p += __shfl_xor(p, off, 32);
            if (lane == 0) {
                int s = gidx[r];
                atomicAdd(&gsum[s], p);
                atomicAdd(&gcnt[s], 1.0f);
            }
        }
    }

    __syncthreads();
    for (int i = tid; i < BINS; i += TPB) {
        float c = s_cnt[i];
        if (c != 0.0f) {
            atomicAdd(&gsum[i], s_sum[i]);
            atomicAdd(&gcnt[i], c);
        }
    }
}

__global__ void gin_finalize(const float* __restrict__ gsum,
                             const float* __restrict__ gcnt,
                             const float* __restrict__ bptr,
                             float* __restrict__ out, int B) {
    int i = blockIdx.x * blockDim.x + threadIdx.x;
    if (i < B) {
        float c = gcnt[i];
        float m = (c > 0.5f) ? (gsum[i] / c) : 0.0f;   // empty segments -> 0
        float z = m + bptr[0];
        out[i] = 1.0f / (1.0f + __expf(-z));
    }
}

extern "C" void kernel_launch(void* const* d_in, const int* in_sizes, int n_in,
                              void* d_out, int out_size, void* d_ws, size_t ws_size,
                              hipStream_t stream) {
    // inputs: 0 nodes [N,256] f32 | 5 node_graph_idx [N] i32 | 7 W [256,1] f32 | 8 b [1] f32
    const float* nodes = (const float*)d_in[0];
    const int*   gidx  = (const int*)  d_in[5];
    const float* W     = (const float*)d_in[7];
    const float* b     = (const float*)d_in[8];

    const int N = in_sizes[0] / DIMS;
    const int B = out_size;                  // 1024

    float* gsum = (float*)d_ws;              // [B]
    float* gcnt = gsum + B;                  // [B]

    const size_t smemBytes = (size_t)SMEM_FLOATS * sizeof(float);  // ~288 KB < 320 KB/WGP

    gin_zero_ws<<<(2 * B + 255) / 256, 256, 0, stream>>>(gsum, 2 * B);
    gin_seg_dot<<<GRID_MAIN, TPB, smemBytes, stream>>>(nodes, gidx, W, gsum, gcnt, N);
    gin_finalize<<<(B + 255) / 256, 256, 0, stream>>>(gsum, gcnt, b, (float*)d_out, B);
}